// PrototypeBank_50491635532082
// MI455X (gfx1250) — compile-verified
//
#include <hip/hip_runtime.h>
#include <stdint.h>

// Problem constants (match the reference).
#define NUM_CLASSES 1000
#define FEATURE_DIM 256
#define NROWS       500000

#define SCATTER_BLOCKS 8192
#define SCATTER_TPB    256

// ---------------------------------------------------------------------------
// CDNA5 async-memory helpers.
// ---------------------------------------------------------------------------
#if __has_builtin(__builtin_amdgcn_s_wait_asynccnt)
#define WAIT_ASYNCCNT(n) __builtin_amdgcn_s_wait_asynccnt(n)
#else
#define WAIT_ASYNCCNT(n) asm volatile("s_wait_asynccnt %0" ::"i"(n) : "memory")
#endif

// Async DMA of 16 bytes from global memory into LDS (per-lane addresses).
// VDST = VGPR holding the LDS byte address, VADDR = 64-bit global address.
// Tracked by ASYNCcnt; no VGPR data path, no LOADcnt.
__device__ __forceinline__ void async_load16_to_lds(const void* gaddr,
                                                    uint32_t lds_byte_off) {
    uint64_t ga = (uint64_t)gaddr;
    asm volatile("global_load_async_to_lds_b128 %0, %1, off"
                 :
                 : "v"(lds_byte_off), "v"(ga)
                 : "memory");
}

// ---------------------------------------------------------------------------
// Kernel 1: zero the scratch accumulator (C*D sums + C counts).
// ---------------------------------------------------------------------------
__global__ void pb_zero_ws(float* __restrict__ ws, int n) {
    int i = blockIdx.x * blockDim.x + threadIdx.x;
    if (i < n) ws[i] = 0.0f;
}

// ---------------------------------------------------------------------------
// Scatter helper: 4 native f32 atomics into the L2-resident accumulator.
// ---------------------------------------------------------------------------
__device__ __forceinline__ void scatter_row_quad(float4 f, int lbl, int lane,
                                                 float* __restrict__ acc,
                                                 float* __restrict__ cnt) {
    float* dst = acc + (size_t)lbl * FEATURE_DIM + lane * 4;
    atomicAdd(dst + 0, f.x);
    atomicAdd(dst + 1, f.y);
    atomicAdd(dst + 2, f.z);
    atomicAdd(dst + 3, f.w);
    if (lane == 0) {
        atomicAdd(cnt + lbl, 1.0f);
    }
}

// ---------------------------------------------------------------------------
// Kernel 2: scatter-reduce with a double-buffered async global->LDS pipeline.
// 64 threads cooperate on one row; each thread async-stages its own 16B into
// LDS (global_load_async_to_lds_b128), waits on ASYNCcnt, reads it back
// (ds_load_b128) and fires 4 global_atomic_add_f32 into acc[label].
// Self-consumption per thread => wave-level s_wait_asynccnt, no barriers.
// ---------------------------------------------------------------------------
__global__ void pb_scatter_async(const float* __restrict__ features,
                                 const int*   __restrict__ labels,
                                 float* __restrict__ acc,   // [C, D]
                                 float* __restrict__ cnt) { // [C]
    __shared__ float4 stage[2][SCATTER_TPB];   // 2 x 4KB double buffer

    const int tid     = blockIdx.x * blockDim.x + threadIdx.x;
    const int lane    = tid & 63;                       // float4 slot within row
    const int rstride = (gridDim.x * blockDim.x) >> 6;  // rows per grid pass
    const int row0    = tid >> 6;

    // Allocation-relative LDS byte offsets for this thread's two stage slots
    // (low 32 bits of a generic pointer to a __shared__ object).
    const uint32_t lds_off0 = (uint32_t)(size_t)&stage[0][threadIdx.x];
    const uint32_t lds_off1 = (uint32_t)(size_t)&stage[1][threadIdx.x];

    if (row0 >= NROWS) return;  // uniform per wave (32 lanes share one row)

    // Prologue: stage the first row.
    async_load16_to_lds(features + (size_t)row0 * FEATURE_DIM + lane * 4, lds_off0);

    int row = row0;
    int buf = 0;
    // Steady state: issue next stage, wait for the older one, consume, scatter.
    while (row + rstride < NROWS) {
        const int nrow = row + rstride;
        async_load16_to_lds(features + (size_t)nrow * FEATURE_DIM + lane * 4,
                            buf ? lds_off0 : lds_off1);
        WAIT_ASYNCCNT(1);                       // previous stage has landed
        const float4 f = stage[buf][threadIdx.x];
        scatter_row_quad(f, labels[row], lane, acc, cnt);
        buf ^= 1;
        row = nrow;
    }
    // Epilogue: drain the last stage.
    WAIT_ASYNCCNT(0);
    const float4 f = stage[buf][threadIdx.x];
    scatter_row_quad(f, labels[row], lane, acc, cnt);
}

// ---------------------------------------------------------------------------
// Kernel 3: EMA finalize into packed [C, D+1] output.
// out[c, 0..255] = present ? m*proto + (1-m)*seg_sum/max(cnt,1) : proto
// out[c, 256]    = present ? m*count + (1-m)*cnt               : count
// ---------------------------------------------------------------------------
__global__ void pb_finalize(const float* __restrict__ protos,   // [C, D]
                            const float* __restrict__ counts,   // [C]
                            const float* __restrict__ acc,      // [C, D]
                            const float* __restrict__ cnt,      // [C]
                            float* __restrict__ out) {          // [C, D+1]
    const int i = blockIdx.x * blockDim.x + threadIdx.x;
    const int total = NUM_CLASSES * (FEATURE_DIM + 1);
    if (i >= total) return;

    const int c = i / (FEATURE_DIM + 1);
    const int j = i - c * (FEATURE_DIM + 1);

    const float n = cnt[c];
    const bool present = n > 0.0f;
    const float m  = 0.9f;
    const float om = 1.0f - m;   // matches f32(1.0) - f32(0.9) in the reference

    if (j < FEATURE_DIM) {
        const float p = protos[c * FEATURE_DIM + j];
        const float mean = acc[c * FEATURE_DIM + j] / fmaxf(n, 1.0f);
        out[i] = present ? (m * p + om * mean) : p;
    } else {
        const float co = counts[c];
        out[i] = present ? (m * co + om * n) : co;
    }
}

// ---------------------------------------------------------------------------
// Host-side launch.
// d_in order: features [N*D] f32, prototypes [C*D] f32, counts [C] f32,
//             labels [N] int.
// d_out: [C*(D+1)] f32. d_ws: >= (C*D + C)*4 bytes scratch.
// ---------------------------------------------------------------------------
extern "C" void kernel_launch(void* const* d_in, const int* in_sizes, int n_in,
                              void* d_out, int out_size, void* d_ws, size_t ws_size,
                              hipStream_t stream) {
    const float* features = (const float*)d_in[0];
    const float* protos   = (const float*)d_in[1];
    const float* counts   = (const float*)d_in[2];
    const int*   labels   = (const int*)d_in[3];

    float* acc = (float*)d_ws;                    // [C, D]
    float* cnt = acc + NUM_CLASSES * FEATURE_DIM; // [C]
    float* out = (float*)d_out;

    const int ws_elems = NUM_CLASSES * FEATURE_DIM + NUM_CLASSES;
    pb_zero_ws<<<(ws_elems + 255) / 256, 256, 0, stream>>>(acc, ws_elems);

    // 8192 blocks x 256 threads -> 32768 rows per grid pass,
    // ~16 grid-stride iterations over the 500k rows.
    pb_scatter_async<<<SCATTER_BLOCKS, SCATTER_TPB, 0, stream>>>(features, labels,
                                                                 acc, cnt);

    const int out_elems = NUM_CLASSES * (FEATURE_DIM + 1);
    pb_finalize<<<(out_elems + 255) / 256, 256, 0, stream>>>(protos, counts, acc,
                                                             cnt, out);
}